// SparseAttention_89421219103270
// MI455X (gfx1250) — compile-verified
//
#include <hip/hip_runtime.h>
#include <stdint.h>
#include <math.h>

// out = (Q K^T * scale) V == scale * Q @ (K^T V)   (no softmax in the reference!)
// K^T V is 64x64 per head -> total FLOPs ~0.54G, I/O = 32MB -> HBM-bound (~1.4us @ 23.3TB/s).
// Matmuls: V_WMMA_F32_16X16X4_F32 (wave32, fp32 like the reference).
// Data movement: Tensor Data Mover (tensor_load_to_lds) with TENSORcnt pipelining;
// TDM's pad feature gives the bank-conflict-free stride-65 Q layout for free.

typedef __attribute__((ext_vector_type(2))) float        v2f;
typedef __attribute__((ext_vector_type(8))) float        v8f;
typedef __attribute__((ext_vector_type(4))) unsigned int v4u;
typedef __attribute__((ext_vector_type(8))) int          v8i;
typedef __attribute__((ext_vector_type(4))) int          v4i;

#define HEADS  8
#define SEQ    4096
#define DIM    64
#define NCHUNK 16               // phase-1 seq chunks: 8*16 = 128 blocks
#define CHUNK  (SEQ / NCHUNK)   // 256 rows per chunk
#define TTILE  64               // K/V rows per LDS tile (ping-pong)
#define ROWS2  128              // Q rows per block in phase 2

// Low 32 bits of a generic LDS pointer = LDS byte offset (flat aperture rule).
__device__ __forceinline__ uint32_t lds_off(const void* p) {
    return (uint32_t)(uintptr_t)p;
}

// Issue one TDM 2D tile load: rows x cols f32, row stride in elements.
// pad1: insert 1 DWORD of LDS padding after every 64 DWORDs (stride 65 layout).
// D# packing per CDNA5 ISA ch.8 (group0: count/lds/global/type, group1: dims).
// This toolchain exposes the 6-arg builtin: (v4u, v8i, v4i, v4i, v8i, i32 cpol).
__device__ __forceinline__ void tdm_load_2d(const void* gptr, uint32_t ldsoff,
                                            uint32_t rows, uint32_t cols,
                                            uint32_t row_stride_elems, bool pad1) {
    const uint64_t ga = (uint64_t)(uintptr_t)gptr;
    v4u g0;
    g0[0] = 1u;                                                // count=1 (valid), user D#
    g0[1] = ldsoff;                                            // lds_addr [63:32]
    g0[2] = (uint32_t)ga;                                      // global_addr lo
    g0[3] = (uint32_t)((ga >> 32) & 0x01FFFFFFu) | 0x80000000u;// ga[56:32] | type=2
    uint32_t w0 = 0x00020000u;                                 // data_size=2 (4 bytes)
    if (pad1) w0 |= (1u << 20) | (5u << 22);                   // pad_en, interval=64dw, amount=1dw
    v8i g1;
    g1[0] = (int)w0;                                           // wg_mask=0, flags
    g1[1] = (int)((cols & 0xFFFFu) << 16);                     // tensor_dim0 lo16
    g1[2] = (int)(((cols >> 16) & 0xFFFFu) | ((rows & 0xFFFFu) << 16)); // dim0 hi / dim1 lo
    g1[3] = (int)(((rows >> 16) & 0xFFFFu) | ((cols & 0xFFFFu) << 16)); // dim1 hi / tile_dim0
    g1[4] = (int)(rows & 0xFFFFu);                             // tile_dim1 ; tile_dim2=0
    g1[5] = (int)row_stride_elems;                             // tensor_dim0_stride lo32
    g1[6] = 0;                                                 // stride hi / dim1_stride lo
    g1[7] = 0;                                                 // dim1_stride hi
    v4i g2 = (v4i)0;                                           // dims 2/3 unused
    v4i g3 = (v4i)0;
    v8i g4 = (v8i)0;                                           // extra group (zeroed)
    __builtin_amdgcn_tensor_load_to_lds(g0, g1, g2, g3, g4, 0);
}

// ---------------- Phase 1: partial[h][c] = K[h][chunk]^T @ V[h][chunk] (64x64) ----
__global__ __launch_bounds__(512)
void ktv_partial_kernel(const float* __restrict__ K, const float* __restrict__ V,
                        float* __restrict__ partial) {
    const int h = blockIdx.x / NCHUNK;
    const int c = blockIdx.x % NCHUNK;
    const float* __restrict__ Kh = K + (size_t)h * SEQ * DIM;
    const float* __restrict__ Vh = V + (size_t)h * SEQ * DIM;

    __shared__ __align__(16) float ldsK[2][TTILE * DIM];   // 2 x 16 KB ping-pong
    __shared__ __align__(16) float ldsV[2][TTILE * DIM];   // 2 x 16 KB

    const int tid  = threadIdx.x;
    const int wave = tid >> 5;            // 16 waves: one 16x16 tile of M each
    const int lane = tid & 31;
    const int l16  = lane & 15;
    const int sel  = lane >> 4;           // 0: K-dims {0,1}, 1: {2,3}
    const int d1   = (wave >> 2) * 16;    // M row tile (K column block)
    const int d2   = (wave &  3) * 16;    // M col tile (V column block)

    const int t_begin = c * CHUNK;
    const int NT = CHUNK / TTILE;         // 4 tiles

    if (wave == 0) {                      // prime the pipeline: tile 0 in flight
        tdm_load_2d(Kh + (size_t)t_begin * DIM, lds_off(&ldsK[0][0]), TTILE, DIM, DIM, false);
        tdm_load_2d(Vh + (size_t)t_begin * DIM, lds_off(&ldsV[0][0]), TTILE, DIM, DIM, false);
    }

    v8f acc = {};
    for (int t = 0; t < NT; ++t) {
        if (wave == 0) {
            if (t + 1 < NT) {             // issue next tile, then wait for current:
                const size_t g = (size_t)(t_begin + (t + 1) * TTILE) * DIM;
                tdm_load_2d(Kh + g, lds_off(&ldsK[(t + 1) & 1][0]), TTILE, DIM, DIM, false);
                tdm_load_2d(Vh + g, lds_off(&ldsV[(t + 1) & 1][0]), TTILE, DIM, DIM, false);
                __builtin_amdgcn_s_wait_tensorcnt((short)2);  // in-order: tile t landed
            } else {
                __builtin_amdgcn_s_wait_tensorcnt((short)0);
            }
        }
        __syncthreads();                  // publish TDM-written LDS to all waves

        const float* bk = &ldsK[t & 1][0];
        const float* bv = &ldsV[t & 1][0];
        #pragma unroll
        for (int ks = 0; ks < TTILE; ks += 4) {
            const int r = ks + sel * 2;
            v2f a, b;
            a.x = bk[(r    ) * DIM + d1 + l16];   // A[m][kk] = K[t..][d1+m]
            a.y = bk[(r + 1) * DIM + d1 + l16];
            b.x = bv[(r    ) * DIM + d2 + l16];   // B[kk][n] = V[t..][d2+n]
            b.y = bv[(r + 1) * DIM + d2 + l16];
            acc = __builtin_amdgcn_wmma_f32_16x16x4_f32(false, a, false, b,
                                                        (short)0, acc, false, false);
        }
        __syncthreads();                  // tile consumed; buffer may be re-filled
    }

    float* __restrict__ P = partial + (size_t)(h * NCHUNK + c) * (DIM * DIM);
    #pragma unroll
    for (int i = 0; i < 8; ++i) {
        const int m = d1 + i + 8 * sel;   // C/D layout: VGPR i -> rows i / i+8
        P[m * DIM + d2 + l16] = acc[i];
    }
}

// ---------------- Phase 1b: M[h] = scale * sum_c partial[h][c]  (deterministic) ---
__global__ __launch_bounds__(256)
void reduce_kernel(const float* __restrict__ partial, float* __restrict__ M,
                   float scale) {
    const int idx = blockIdx.x * 256 + threadIdx.x;
    const int h = idx / (DIM * DIM);
    const int e = idx % (DIM * DIM);
    float s = 0.0f;
    #pragma unroll
    for (int c = 0; c < NCHUNK; ++c)
        s += partial[(size_t)(h * NCHUNK + c) * (DIM * DIM) + e];
    M[idx] = s * scale;
}

// ---------------- Phase 2: out[h] = Q[h] @ M[h]  ([4096x64] @ [64x64]) ------------
__global__ __launch_bounds__(256)
void qm_kernel(const float* __restrict__ Q, const float* __restrict__ M,
               float* __restrict__ out) {
    const int nblk = SEQ / ROWS2;          // 32
    const int h  = blockIdx.x / nblk;
    const int rb = (blockIdx.x % nblk) * ROWS2;
    const float* __restrict__ Qh = Q   + (size_t)h * SEQ * DIM;
    const float* __restrict__ Mh = M   + (size_t)h * DIM * DIM;
    float*       __restrict__ Oh = out + (size_t)h * SEQ * DIM;

    const int QS = DIM + 1;                // TDM pad -> stride 65, no bank conflicts
    __shared__ __align__(16) float ldsQ[ROWS2 * (DIM + 1)];  // ~33 KB
    __shared__ __align__(16) float ldsM[DIM * DIM];          // 16 KB

    const int tid  = threadIdx.x;
    const int wave = tid >> 5;             // 8 waves: 16 rows x 64 cols each
    const int lane = tid & 31;
    const int l16  = lane & 15;
    const int sel  = lane >> 4;
    (void)tid;

    if (wave == 0) {                       // TDM: Q tile (padded) + M tile
        tdm_load_2d(Qh + (size_t)rb * DIM, lds_off(&ldsQ[0]), ROWS2, DIM, DIM, true);
        tdm_load_2d(Mh,                    lds_off(&ldsM[0]), DIM,   DIM, DIM, false);
        __builtin_amdgcn_s_wait_tensorcnt((short)0);
    }
    __syncthreads();

    const int wr = wave * 16;
    v8f acc0 = {}, acc1 = {}, acc2 = {}, acc3 = {};
    #pragma unroll
    for (int k0 = 0; k0 < DIM; k0 += 4) {
        const int r = k0 + sel * 2;
        v2f a;                              // A[m][kk] = Q[rb+wr+m][r+kk]
        a.x = ldsQ[(wr + l16) * QS + r];
        a.y = ldsQ[(wr + l16) * QS + r + 1];
        v2f b0, b1, b2, b3;                 // B[kk][n] = M[r+kk][j*16+n]
        b0.x = ldsM[(r) * DIM +  0 + l16];  b0.y = ldsM[(r + 1) * DIM +  0 + l16];
        b1.x = ldsM[(r) * DIM + 16 + l16];  b1.y = ldsM[(r + 1) * DIM + 16 + l16];
        b2.x = ldsM[(r) * DIM + 32 + l16];  b2.y = ldsM[(r + 1) * DIM + 32 + l16];
        b3.x = ldsM[(r) * DIM + 48 + l16];  b3.y = ldsM[(r + 1) * DIM + 48 + l16];
        acc0 = __builtin_amdgcn_wmma_f32_16x16x4_f32(false, a, false, b0, (short)0, acc0, false, false);
        acc1 = __builtin_amdgcn_wmma_f32_16x16x4_f32(false, a, false, b1, (short)0, acc1, false, false);
        acc2 = __builtin_amdgcn_wmma_f32_16x16x4_f32(false, a, false, b2, (short)0, acc2, false, false);
        acc3 = __builtin_amdgcn_wmma_f32_16x16x4_f32(false, a, false, b3, (short)0, acc3, false, false);
    }

    #pragma unroll
    for (int i = 0; i < 8; ++i) {
        const int row = rb + wr + i + 8 * sel;
        float* orow = Oh + (size_t)row * DIM;
        orow[ 0 + l16] = acc0[i];
        orow[16 + l16] = acc1[i];
        orow[32 + l16] = acc2[i];
        orow[48 + l16] = acc3[i];
    }
}

// ----------------------------------------------------------------------------------
extern "C" void kernel_launch(void* const* d_in, const int* in_sizes, int n_in,
                              void* d_out, int out_size, void* d_ws, size_t ws_size,
                              hipStream_t stream) {
    (void)in_sizes; (void)n_in; (void)out_size; (void)ws_size;
    const float* q = (const float*)d_in[0];
    const float* k = (const float*)d_in[1];
    const float* v = (const float*)d_in[2];
    float* out = (float*)d_out;

    // ws: [HEADS*NCHUNK][64*64] partials, then [HEADS][64*64] reduced M (~2.2 MB)
    float* partial = (float*)d_ws;
    float* M       = partial + (size_t)HEADS * NCHUNK * DIM * DIM;

    const float scale = 1.0f / sqrtf((float)DIM);

    ktv_partial_kernel<<<HEADS * NCHUNK, 512, 0, stream>>>(k, v, partial);
    reduce_kernel<<<(HEADS * DIM * DIM) / 256, 256, 0, stream>>>(partial, M, scale);
    qm_kernel<<<HEADS * (SEQ / ROWS2), 256, 0, stream>>>(q, M, out);
}